// DotPredictor_54692113547268
// MI455X (gfx1250) — compile-verified
//
#include <hip/hip_runtime.h>
#include <hip/hip_bf16.h>
#include <math.h>

// u_dot_v + sigmoid for 64-dim f32 embeddings on gfx1250 (MI455X).
//
// Roofline: 102 MFLOP vs ~410MB of gathered rows; h (12.8MB) is L2-resident
// (192MB L2), so the bound is L2 request throughput. Therefore:
//   * widest possible gather: each wave stages 32 rows (16 src + 16 dst)
//     with float4 (global_load_b128) — 16 lanes per 256B row, one load
//     instruction stages two rows. 16B/lane requests, fully coalesced.
//   * wave-private padded LDS tile (stride 68 floats): b128 stores hit all
//     64 banks exactly twice (minimum); b64 WMMA-fragment reads are
//     conflict-free.
//   * V_WMMA_F32_16X16X4_F32 accumulated over 16 K-chunks computes the
//     16x16 src×dst cross-product; the diagonal is the 16 per-edge dots.
//     f32 end-to-end matches the reference; the K-reduction rides the
//     matrix pipe (free — memory bound), no cross-lane shuffles.
//   * diagonal extraction via unrolled selects, sigmoid, predicated store.

typedef __attribute__((ext_vector_type(2))) float v2f;
typedef __attribute__((ext_vector_type(8))) float v8f;

#define WAVES_PER_BLOCK 4
#define BLOCK_THREADS   (WAVES_PER_BLOCK * 32)
#define ROW_STRIDE      68                  // 64 data floats + 4 pad (272B, 16B-aligned)
#define LDS_PER_WAVE    (32 * ROW_STRIDE)   // 32 rows

__global__ __launch_bounds__(BLOCK_THREADS)
void DotPredictor_udotv_sigmoid(const float* __restrict__ h,
                                const int*   __restrict__ src,
                                const int*   __restrict__ dst,
                                float*       __restrict__ out,
                                int E)
{
    __shared__ __align__(16) float lds[WAVES_PER_BLOCK * LDS_PER_WAVE];

    const int lane = threadIdx.x & 31;
    const int wave = threadIdx.x >> 5;
    float* myLds = lds + wave * LDS_PER_WAVE;

    const int m    = lane & 15;   // row-within-pair / edge-in-batch id
    const int half = lane >> 4;   // staging: row select; WMMA: K=0,1 vs K=2,3

    const int nBatches  = (E + 15) >> 4;
    const int waveGid   = blockIdx.x * WAVES_PER_BLOCK + wave;
    const int waveCount = gridDim.x * WAVES_PER_BLOCK;

    for (int b = waveGid; b < nBatches; b += waveCount) {
        const int e0 = b << 4;

        // Each half-wave redundantly holds the 16 edge indices of this batch
        // (clamped for a possible ragged tail; stores are exactly guarded).
        int ei = e0 + m;
        if (ei >= E) ei = E - 1;
        const int sIdx = src[ei];
        const int dIdx = dst[ei];

        // ---- stage 32 rows into LDS: one b128 load covers two rows ----
        // lanes 0-15 -> row r (16 x float4 = 256B), lanes 16-31 -> row r+1
        #pragma unroll
        for (int r = 0; r < 16; r += 2) {
            int nA = __builtin_amdgcn_readlane(sIdx, r);
            int nB = __builtin_amdgcn_readlane(sIdx, r + 1);
            int node = half ? nB : nA;
            const float4* rp = (const float4*)(h + (long)node * 64);
            float4 v = rp[m];                              // global_load_b128
            *(float4*)(myLds + (r + half) * ROW_STRIDE + 4 * m) = v;  // ds_store_b128
        }
        #pragma unroll
        for (int r = 0; r < 16; r += 2) {
            int nA = __builtin_amdgcn_readlane(dIdx, r);
            int nB = __builtin_amdgcn_readlane(dIdx, r + 1);
            int node = half ? nB : nA;
            const float4* rp = (const float4*)(h + (long)node * 64);
            float4 v = rp[m];
            *(float4*)(myLds + (16 + r + half) * ROW_STRIDE + 4 * m) = v;
        }
        // Wave-private staging: same-wave LDS ops are ordered, no barrier needed.

        // ---- 16x WMMA f32 16x16x4: D = A(src rows) x B(dst cols) + C ----
        // A/B f32 fragment layout: lanes 0-15 carry K=0,1; lanes 16-31 K=2,3.
        const float* aBase = myLds + m * ROW_STRIDE + half * 2;
        const float* bBase = myLds + (16 + m) * ROW_STRIDE + half * 2;

        v8f c = {};
        #pragma unroll
        for (int k = 0; k < 16; ++k) {
            v2f a  = *(const v2f*)(aBase + 4 * k);   // ds_load_b64, conflict-free
            v2f bb = *(const v2f*)(bBase + 4 * k);
            c = __builtin_amdgcn_wmma_f32_16x16x4_f32(
                    /*neg_a=*/false, a, /*neg_b=*/false, bb,
                    /*c_mod=*/(short)0, c, /*reuse_a=*/false, /*reuse_b=*/false);
        }

        // ---- diagonal extraction: edge i (i<8) at (vgpr i, lane i),
        //      edge i (i>=8) at (vgpr i-8, lane 16+i) i.e. lanes 24..31 ----
        float s = c[0];
        #pragma unroll
        for (int v = 1; v < 8; ++v) {
            bool sel = (lane == v) | (lane == 24 + v);
            s = sel ? c[v] : s;
        }

        if (lane < 8 || lane >= 24) {
            int edge = (lane < 8) ? lane : (lane - 16);   // 0..15
            int eo = e0 + edge;
            if (eo < E) out[eo] = 1.0f / (1.0f + __expf(-s));
        }
    }
}

extern "C" void kernel_launch(void* const* d_in, const int* in_sizes, int n_in,
                              void* d_out, int out_size, void* d_ws, size_t ws_size,
                              hipStream_t stream)
{
    const float* h   = (const float*)d_in[0];
    const int*   src = (const int*)d_in[1];   // harness contract: integer -> const int*
    const int*   dst = (const int*)d_in[2];
    float*       out = (float*)d_out;

    const int E = in_sizes[1];                // edge count
    const int nBatches = (E + 15) >> 4;

    int grid = (nBatches + WAVES_PER_BLOCK - 1) / WAVES_PER_BLOCK;
    if (grid > 32768) grid = 32768;
    if (grid < 1)     grid = 1;

    hipLaunchKernelGGL(DotPredictor_udotv_sigmoid,
                       dim3(grid), dim3(BLOCK_THREADS), 0, stream,
                       h, src, dst, out, E);
}